// ProbAttention_10144712753264
// MI455X (gfx1250) — compile-verified
//
#include <hip/hip_runtime.h>
#include <hip/hip_bf16.h>

typedef __attribute__((ext_vector_type(2))) float v2f;
typedef __attribute__((ext_vector_type(8))) float v8f;

#define Bdim 4
#define Ldim 2048
#define Hdim 8
#define Ddim 64
#define SK   40     // sample_k
#define NT   40     // n_top
#define NTP  48     // n_top padded to 3 WMMA M-tiles
#define BH   (Bdim*Hdim)
#define QSTR 68     // LDS row stride for Q_reduce (bank-conflict pad)

// ---------------------------------------------------------------------------
// 1) M[b,h,l] = max_s(Q.Ks) - sum_s(Q.Ks)/L   over 40 sampled keys
// ---------------------------------------------------------------------------
__global__ void k_compute_m(const float* __restrict__ Q, const float* __restrict__ K,
                            const int* __restrict__ idx, float* __restrict__ Mout) {
    int t  = blockIdx.x * blockDim.x + threadIdx.x;   // t = bh*L + l, exact grid
    int l  = t & (Ldim - 1);
    int bh = t >> 11;
    int b  = bh / Hdim, h = bh % Hdim;

    const float4* q = (const float4*)(Q + ((size_t)(b * Ldim + l) * Hdim + h) * Ddim);
    float4 qv[16];
#pragma unroll
    for (int i = 0; i < 16; ++i) qv[i] = q[i];

    float mx = -3.4e38f, sm = 0.f;
    for (int s = 0; s < SK; ++s) {
        int ki = idx[l * SK + s];
        const float4* kp = (const float4*)(K + ((size_t)(b * Ldim + ki) * Hdim + h) * Ddim);
        float acc = 0.f;
#pragma unroll
        for (int i = 0; i < 16; ++i) {
            float4 kv = kp[i];
            acc += qv[i].x * kv.x + qv[i].y * kv.y + qv[i].z * kv.z + qv[i].w * kv.w;
        }
        mx = fmaxf(mx, acc);
        sm += acc;
    }
    Mout[t] = mx - sm * (1.0f / (float)Ldim);
}

// ---------------------------------------------------------------------------
// 2) top-40 indices per (b,h): iterative argmax (ties -> smaller index)
// ---------------------------------------------------------------------------
__global__ void k_topk(const float* __restrict__ Mv, int* __restrict__ top) {
    __shared__ float vals[Ldim];
    __shared__ float redv[256];
    __shared__ int   redi[256];
    int bh = blockIdx.x, tid = threadIdx.x;

    for (int i = tid; i < Ldim; i += 256) vals[i] = Mv[(size_t)bh * Ldim + i];
    __syncthreads();

    for (int it = 0; it < NT; ++it) {
        float bv = -3.4e38f; int bi = Ldim;
        for (int i = tid; i < Ldim; i += 256) {
            float v = vals[i];
            if (v > bv || (v == bv && i < bi)) { bv = v; bi = i; }
        }
        redv[tid] = bv; redi[tid] = bi;
        __syncthreads();
        for (int s = 128; s > 0; s >>= 1) {
            if (tid < s) {
                float v2 = redv[tid + s]; int i2 = redi[tid + s];
                if (v2 > redv[tid] || (v2 == redv[tid] && i2 < redi[tid])) {
                    redv[tid] = v2; redi[tid] = i2;
                }
            }
            __syncthreads();
        }
        if (tid == 0) { top[bh * NT + it] = redi[0]; vals[redi[0]] = -3.4e38f; }
        __syncthreads();
    }
}

// ---------------------------------------------------------------------------
// 3a) V mean over L per (b,h,d)    3b) broadcast mean into output
// ---------------------------------------------------------------------------
__global__ void k_vmean(const float* __restrict__ V, float* __restrict__ mean) {
    int bh = blockIdx.x, d = threadIdx.x;          // 64 threads
    int b = bh / Hdim, h = bh % Hdim;
    float s = 0.f;
    for (int l = 0; l < Ldim; ++l)
        s += V[((size_t)(b * Ldim + l) * Hdim + h) * Ddim + d];
    mean[bh * Ddim + d] = s * (1.0f / (float)Ldim);
}

__global__ void k_fill(const float* __restrict__ mean, float* __restrict__ out) {
    size_t t = (size_t)blockIdx.x * 256 + threadIdx.x;  // enumerates (bh,l,d)
    int d  = (int)(t % Ddim);
    int bh = (int)(t / ((size_t)Ddim * Ldim));
    out[t] = mean[bh * Ddim + d];
}

// ---------------------------------------------------------------------------
// 4a) scores[bh][0..47][0..2047] = scale * Q_reduce @ K^T   (WMMA f32 16x16x4)
//     grid (32 col-slabs of 64, BH), block 128 = 4 waves, wave owns one N-tile
// ---------------------------------------------------------------------------
__global__ void k_scores(const float* __restrict__ Q, const float* __restrict__ K,
                         const int* __restrict__ top, float* __restrict__ scores) {
    __shared__ float Qs[NTP * QSTR];
    int bh = blockIdx.y;
    int b = bh / Hdim, h = bh % Hdim;
    int n0 = blockIdx.x * 64;
    int tid = threadIdx.x;

    // gather Q_reduce (rows >= 40 zeroed); 3072/128 = 24 iters, no divergence
    for (int i = tid; i < NTP * Ddim; i += 128) {
        int r = i >> 6, d = i & 63;
        float v = 0.f;
        if (r < NT) {
            int l = top[bh * NT + r];
            v = Q[((size_t)(b * Ldim + l) * Hdim + h) * Ddim + d];
        }
        Qs[r * QSTR + d] = v;
    }
    __syncthreads();

    int wave = tid >> 5, lane = tid & 31;
    int nloc = lane & 15;
    int koff = (lane >= 16) ? 2 : 0;
    int col  = n0 + wave * 16 + nloc;

    const float* kbase = K + ((size_t)(b * Ldim + col) * Hdim + h) * Ddim;

    v8f acc0 = {}, acc1 = {}, acc2 = {};
    for (int k0 = 0; k0 < Ddim; k0 += 4) {
        v2f bv = *(const v2f*)(kbase + k0 + koff);            // B[k][n] = K[col][k]
        v2f a0 = *(const v2f*)&Qs[(nloc)      * QSTR + k0 + koff];
        v2f a1 = *(const v2f*)&Qs[(16 + nloc) * QSTR + k0 + koff];
        v2f a2 = *(const v2f*)&Qs[(32 + nloc) * QSTR + k0 + koff];
        acc0 = __builtin_amdgcn_wmma_f32_16x16x4_f32(false, a0, false, bv, (short)0, acc0, false, false);
        acc1 = __builtin_amdgcn_wmma_f32_16x16x4_f32(false, a1, false, bv, (short)0, acc1, false, false);
        acc2 = __builtin_amdgcn_wmma_f32_16x16x4_f32(false, a2, false, bv, (short)0, acc2, false, false);
    }

    const float scale = 0.125f;                               // 1/sqrt(64)
    int rbase = (lane >= 16) ? 8 : 0;
#pragma unroll
    for (int r = 0; r < 8; ++r) {
        int row = r + rbase;
        scores[((size_t)bh * NTP + row)      * Ldim + col] = acc0[r] * scale;
        scores[((size_t)bh * NTP + 16 + row) * Ldim + col] = acc1[r] * scale;
        scores[((size_t)bh * NTP + 32 + row) * Ldim + col] = acc2[r] * scale;
    }
}

// ---------------------------------------------------------------------------
// 4b) row softmax over 2048 (one block per padded row)
// ---------------------------------------------------------------------------
__global__ void k_softmax(float* __restrict__ scores) {
    __shared__ float red[256];
    int tid = threadIdx.x;
    float* p = scores + (size_t)blockIdx.x * Ldim;

    float m = -3.4e38f;
    for (int i = tid; i < Ldim; i += 256) m = fmaxf(m, p[i]);
    red[tid] = m; __syncthreads();
    for (int s = 128; s > 0; s >>= 1) { if (tid < s) red[tid] = fmaxf(red[tid], red[tid + s]); __syncthreads(); }
    m = red[0]; __syncthreads();

    float sum = 0.f;
    for (int i = tid; i < Ldim; i += 256) { float e = __expf(p[i] - m); p[i] = e; sum += e; }
    red[tid] = sum; __syncthreads();
    for (int s = 128; s > 0; s >>= 1) { if (tid < s) red[tid] += red[tid + s]; __syncthreads(); }
    float inv = 1.0f / red[0];
    __syncthreads();
    for (int i = tid; i < Ldim; i += 256) p[i] *= inv;
}

// ---------------------------------------------------------------------------
// 4c) out_sel = attn @ V, scatter rows by M_top   (WMMA f32 16x16x4)
//     one block (4 waves) per (b,h); wave owns a 16-wide D-tile
// ---------------------------------------------------------------------------
__global__ void k_av(const float* __restrict__ scores, const float* __restrict__ V,
                     const int* __restrict__ top, float* __restrict__ out) {
    int bh = blockIdx.x;
    int b = bh / Hdim, h = bh % Hdim;
    int tid = threadIdx.x, wave = tid >> 5, lane = tid & 31;
    int nloc = lane & 15;
    int koff = (lane >= 16) ? 2 : 0;
    int d0 = wave * 16;

    const float* srow = scores + ((size_t)bh * NTP + nloc) * Ldim;
    const float* vb   = V + (((size_t)b * Ldim) * Hdim + h) * Ddim + d0 + nloc;

    v8f acc0 = {}, acc1 = {}, acc2 = {};
    for (int k0 = 0; k0 < Ldim; k0 += 4) {
        int kk = k0 + koff;
        v2f bv = { vb[(size_t)kk * (Hdim * Ddim)],
                   vb[(size_t)(kk + 1) * (Hdim * Ddim)] };   // B[k][n] = V[kk][d0+n]
        v2f a0 = *(const v2f*)(srow + kk);
        v2f a1 = *(const v2f*)(srow + 16 * Ldim + kk);
        v2f a2 = *(const v2f*)(srow + 32 * Ldim + kk);
        acc0 = __builtin_amdgcn_wmma_f32_16x16x4_f32(false, a0, false, bv, (short)0, acc0, false, false);
        acc1 = __builtin_amdgcn_wmma_f32_16x16x4_f32(false, a1, false, bv, (short)0, acc1, false, false);
        acc2 = __builtin_amdgcn_wmma_f32_16x16x4_f32(false, a2, false, bv, (short)0, acc2, false, false);
    }

    int rbase = (lane >= 16) ? 8 : 0;
#pragma unroll
    for (int r = 0; r < 8; ++r) {
        int row0 = r + rbase;
        {   int l = top[bh * NT + row0];                 // rows 0..15 all valid
            out[((size_t)bh * Ldim + l) * Ddim + d0 + nloc] = acc0[r]; }
        {   int row = 16 + row0;
            int l = top[bh * NT + row];                  // rows 16..31 all valid
            out[((size_t)bh * Ldim + l) * Ddim + d0 + nloc] = acc1[r]; }
        {   int row = 32 + row0;
            if (row < NT) {                              // rows 32..39 valid, 40..47 padding
                int l = top[bh * NT + row];
                out[((size_t)bh * Ldim + l) * Ddim + d0 + nloc] = acc2[r];
            } }
    }
}

// ---------------------------------------------------------------------------
extern "C" void kernel_launch(void* const* d_in, const int* in_sizes, int n_in,
                              void* d_out, int out_size, void* d_ws, size_t ws_size,
                              hipStream_t stream) {
    const float* Q   = (const float*)d_in[0];
    const float* K   = (const float*)d_in[1];
    const float* V   = (const float*)d_in[2];
    // d_in[3] = attn_mask (unused, zeros)
    const int*   idx = (const int*)d_in[4];
    float* out = (float*)d_out;

    char* ws = (char*)d_ws;
    float* ws_M      = (float*)(ws);                     // BH*L floats    = 256 KB
    int*   ws_top    = (int*)  (ws + 256 * 1024);        // BH*NT ints     (8 KB slot)
    float* ws_mean   = (float*)(ws + 264 * 1024);        // BH*D floats    (8 KB slot)
    float* ws_scores = (float*)(ws + 272 * 1024);        // BH*48*2048 fl  = 12.6 MB

    k_compute_m<<<(BH * Ldim) / 256, 256, 0, stream>>>(Q, K, idx, ws_M);
    k_topk     <<<BH, 256, 0, stream>>>(ws_M, ws_top);
    k_vmean    <<<BH, Ddim, 0, stream>>>(V, ws_mean);
    k_fill     <<<(BH * Ldim * Ddim) / 256, 256, 0, stream>>>(ws_mean, out);
    k_scores   <<<dim3(Ldim / 64, BH), 128, 0, stream>>>(Q, K, ws_top, ws_scores);
    k_softmax  <<<BH * NTP, 256, 0, stream>>>(ws_scores);
    k_av       <<<BH, 128, 0, stream>>>(ws_scores, V, ws_top, out);
}